// GAT_68143951118601
// MI455X (gfx1250) — compile-verified
//
#include <hip/hip_runtime.h>
#include <hip/hip_bf16.h>

// ---------------------------------------------------------------------------
// 2-layer GATConv (PyG semantics) for MI455X / gfx1250.
// GEMMs: V_WMMA_F32_16X16X4_F32, wave computes a 16x64 strip (4 accumulators,
// A-fragment reused 4x, pointer-bumped addressing).
// Edge softmax/aggregation: f32 global atomics (L2 atomic units).
// ---------------------------------------------------------------------------

typedef float v8f __attribute__((ext_vector_type(8)));
typedef float v2f __attribute__((ext_vector_type(2)));

static __device__ __forceinline__ float atomicMaxFloat(float* addr, float val) {
    // Works given buffer initialized to -inf.
    if (val >= 0.0f) {
        return __int_as_float(atomicMax((int*)addr, __float_as_int(val)));
    } else {
        return __uint_as_float(atomicMin((unsigned int*)addr, __float_as_uint(val)));
    }
}

#define IN_DIM   128
#define HID_DIM  1024
#define OUT_DIM  128
#define H1c      4
#define C1c      64
#define H2c      1
#define C2c      128

// ---------------------------------------------------------------------------
__global__ void __launch_bounds__(256)
fill_f32(float* __restrict__ p, float v, int n) {
    int t = blockIdx.x * blockDim.x + threadIdx.x;
    if (t < n) p[t] = v;
}

// ---------------------------------------------------------------------------
// GEMM: C[M,Nc] = A[M,K] * B[K,Nc]  (row major, M%16==0, K%4==0, Nc%64==0)
// One wave computes a 16x64 strip: 4x V_WMMA_F32_16X16X4_F32 per k-step,
// A fragment loaded once and reused for all 4 column blocks.
// Fragment layouts per CDNA5 ISA 7.12.2 (32-bit A 16x4, B 4x16, C/D 16x16).
// ---------------------------------------------------------------------------
__global__ void __launch_bounds__(256)
gemm_wmma_f32_nb4(const float* __restrict__ A, const float* __restrict__ B,
                  float* __restrict__ C, int M, int K, int Nc) {
    const int wave = (blockIdx.x * blockDim.x + threadIdx.x) >> 5;
    const int lane = threadIdx.x & 31;
    const int nBlk = Nc >> 6;                  // 64-column blocks
    const int tiles = (M >> 4) * nBlk;
    if (wave >= tiles) return;                 // wave-uniform: EXEC stays full

    const int mt  = wave / nBlk;
    const int nb  = wave % nBlk;
    const int hi  = lane >> 4;                 // lane half
    const int l15 = lane & 15;

    // A: lane holds row (lane&15); VGPR0 K = 2*hi, VGPR1 K = 2*hi+1
    const float* Ap = A + (size_t)(mt * 16 + l15) * K + hi * 2;
    // B: lane holds column (nb*64 + j*16 + (lane&15)); rows 2*hi, 2*hi+1
    const float* Bp = B + (size_t)(hi * 2) * Nc + nb * 64 + l15;

    v8f acc0 = {}, acc1 = {}, acc2 = {}, acc3 = {};
    for (int k0 = 0; k0 < K; k0 += 4) {
        v2f a;
        a.x = Ap[0];
        a.y = Ap[1];
        v2f b0, b1, b2, b3;
        b0.x = Bp[0];  b0.y = Bp[Nc + 0];
        b1.x = Bp[16]; b1.y = Bp[Nc + 16];
        b2.x = Bp[32]; b2.y = Bp[Nc + 32];
        b3.x = Bp[48]; b3.y = Bp[Nc + 48];
        acc0 = __builtin_amdgcn_wmma_f32_16x16x4_f32(false, a, false, b0, (short)0, acc0, false, false);
        acc1 = __builtin_amdgcn_wmma_f32_16x16x4_f32(false, a, false, b1, (short)0, acc1, false, false);
        acc2 = __builtin_amdgcn_wmma_f32_16x16x4_f32(false, a, false, b2, (short)0, acc2, false, false);
        acc3 = __builtin_amdgcn_wmma_f32_16x16x4_f32(false, a, false, b3, (short)0, acc3, false, false);
        Ap += 4;
        Bp += (size_t)4 * Nc;
    }

    // C/D layout: VGPR i -> row (i + 8*hi), col = lane&15 within each 16-block
    float* Cr = C + (size_t)(mt * 16 + hi * 8) * Nc + nb * 64 + l15;
#pragma unroll
    for (int i = 0; i < 8; ++i) {
        Cr[(size_t)i * Nc + 0]  = acc0[i];
        Cr[(size_t)i * Nc + 16] = acc1[i];
        Cr[(size_t)i * Nc + 32] = acc2[i];
        Cr[(size_t)i * Nc + 48] = acc3[i];
    }
}

// ---------------------------------------------------------------------------
// per-(node, head) attention scores: es = <h[n,h,:], a_src[h,:]>, ed likewise
// ---------------------------------------------------------------------------
template <int H, int C>
__global__ void __launch_bounds__(256)
node_scores(const float* __restrict__ h, const float* __restrict__ a_src,
            const float* __restrict__ a_dst, float* __restrict__ es,
            float* __restrict__ ed, int N) {
    int t = blockIdx.x * blockDim.x + threadIdx.x;
    if (t >= N * H) return;
    int n = t / H, hd = t % H;
    const float* hp = h + (size_t)n * H * C + (size_t)hd * C;
    const float* sp = a_src + (size_t)hd * C;
    const float* dp = a_dst + (size_t)hd * C;
    float s = 0.f, d0 = 0.f;
#pragma unroll 4
    for (int c = 0; c < C; ++c) {
        float v = hp[c];
        s  += v * sp[c];
        d0 += v * dp[c];
    }
    es[t] = s;
    ed[t] = d0;
}

static __device__ __forceinline__ void edge_sd(const int* __restrict__ ei, int E,
                                               int e, int& s, int& d) {
    if (e < E) { s = ei[e]; d = ei[E + e]; }
    else       { s = e - E; d = e - E; }     // appended self-loops
}

// ---------------------------------------------------------------------------
// pass 1: segment max of leaky_relu(es[src]+ed[dst]) over dst
// ---------------------------------------------------------------------------
template <int H>
__global__ void __launch_bounds__(256)
edge_max(const int* __restrict__ ei, int E, int ET, const float* __restrict__ es,
         const float* __restrict__ ed, float* __restrict__ m) {
    int t = blockIdx.x * blockDim.x + threadIdx.x;
    if (t >= ET * H) return;
    int e = t / H, hd = t % H;
    int s, d; edge_sd(ei, E, e, s, d);
    float x = es[s * H + hd] + ed[d * H + hd];
    x = (x > 0.f) ? x : 0.2f * x;            // leaky_relu, PyG default slope
    atomicMaxFloat(&m[d * H + hd], x);
}

// ---------------------------------------------------------------------------
// pass 2: p = exp(e - m[dst]); z[dst] += p
// ---------------------------------------------------------------------------
template <int H>
__global__ void __launch_bounds__(256)
edge_expsum(const int* __restrict__ ei, int E, int ET, const float* __restrict__ es,
            const float* __restrict__ ed, const float* __restrict__ m,
            float* __restrict__ p, float* __restrict__ z) {
    int t = blockIdx.x * blockDim.x + threadIdx.x;
    if (t >= ET * H) return;
    int e = t / H, hd = t % H;
    int s, d; edge_sd(ei, E, e, s, d);
    float x = es[s * H + hd] + ed[d * H + hd];
    x = (x > 0.f) ? x : 0.2f * x;
    float pe = __expf(x - m[d * H + hd]);
    p[t] = pe;
    atomicAdd(&z[d * H + hd], pe);
}

// ---------------------------------------------------------------------------
// pass 3: acc[dst, hc] += (p/z) * h[src, hc], 4 channels per thread
// ---------------------------------------------------------------------------
template <int H, int C>
__global__ void __launch_bounds__(256)
edge_aggregate(const int* __restrict__ ei, int E, int ET, const float* __restrict__ h,
               const float* __restrict__ p, const float* __restrict__ z,
               float* __restrict__ acc) {
    constexpr int HC = H * C;
    constexpr int chunks = HC >> 2;
    int t = blockIdx.x * blockDim.x + threadIdx.x;
    if (t >= ET * chunks) return;
    int e = t / chunks, q = t % chunks;
    int hc = q << 2;
    int hd = hc / C;
    int s, d; edge_sd(ei, E, e, s, d);
    float alpha = p[e * H + hd] / z[d * H + hd];
    const float4 v = *(const float4*)(h + (size_t)s * HC + hc);
    float* ap = acc + (size_t)d * HC + hc;
    atomicAdd(ap + 0, alpha * v.x);
    atomicAdd(ap + 1, alpha * v.y);
    atomicAdd(ap + 2, alpha * v.z);
    atomicAdd(ap + 3, alpha * v.w);
}

// ---------------------------------------------------------------------------
// epilogue layer 1: h2 = elu(acc + b1)   (in place on acc)
// ---------------------------------------------------------------------------
__global__ void __launch_bounds__(256)
finish1(float* __restrict__ acc, const float* __restrict__ b, int N, int HC) {
    int t = blockIdx.x * blockDim.x + threadIdx.x;
    if (t >= N * HC) return;
    float v = acc[t] + b[t % HC];
    acc[t] = (v > 0.f) ? v : expm1f(v);      // elu, alpha=1
}

// ---------------------------------------------------------------------------
// epilogue layer 2: out += b2  (mean over 1 head == identity)
// ---------------------------------------------------------------------------
__global__ void __launch_bounds__(256)
finish2(float* __restrict__ out, const float* __restrict__ b, int N, int D) {
    int t = blockIdx.x * blockDim.x + threadIdx.x;
    if (t >= N * D) return;
    out[t] += b[t % D];
}

static inline int cdiv(long a, long b) { return (int)((a + b - 1) / b); }

extern "C" void kernel_launch(void* const* d_in, const int* in_sizes, int n_in,
                              void* d_out, int out_size, void* d_ws, size_t ws_size,
                              hipStream_t stream) {
    const float* x   = (const float*)d_in[0];
    const int*   ei  = (const int*)  d_in[1];
    const float* W1  = (const float*)d_in[2];
    const float* as1 = (const float*)d_in[3];
    const float* ad1 = (const float*)d_in[4];
    const float* b1  = (const float*)d_in[5];
    const float* W2  = (const float*)d_in[6];
    const float* as2 = (const float*)d_in[7];
    const float* ad2 = (const float*)d_in[8];
    const float* b2  = (const float*)d_in[9];

    const int N  = in_sizes[0] / IN_DIM;     // 50000
    const int E  = in_sizes[1] / 2;          // 800000
    const int ET = E + N;                    // + self loops

    // workspace layout (floats)
    float* ws   = (float*)d_ws;
    float* h1   = ws;                                 // N*HID
    float* acc1 = h1   + (size_t)N * HID_DIM;         // N*HID (becomes h2)
    float* t2   = acc1 + (size_t)N * HID_DIM;         // N*OUT
    float* es1  = t2   + (size_t)N * OUT_DIM;         // N*H1
    float* ed1  = es1  + (size_t)N * H1c;
    float* m1   = ed1  + (size_t)N * H1c;
    float* z1   = m1   + (size_t)N * H1c;
    float* es2  = z1   + (size_t)N * H1c;             // N
    float* ed2  = es2  + (size_t)N;
    float* m2   = ed2  + (size_t)N;
    float* z2   = m2   + (size_t)N;
    float* p1   = z2   + (size_t)N;                   // ET*H1
    float* p2   = p1   + (size_t)ET * H1c;            // ET
    float* out  = (float*)d_out;                      // doubles as layer-2 acc

    const int TPB = 256;
    const float NEG_INF = -__builtin_huge_valf();

    // ---------------- layer 1 ----------------
    {
        int waves = (N / 16) * (HID_DIM / 64);        // 16x64 strips
        gemm_wmma_f32_nb4<<<cdiv((long)waves * 32, TPB), TPB, 0, stream>>>(
            x, W1, h1, N, IN_DIM, HID_DIM);
    }
    node_scores<H1c, C1c><<<cdiv((long)N * H1c, TPB), TPB, 0, stream>>>(
        h1, as1, ad1, es1, ed1, N);

    fill_f32<<<cdiv((long)N * H1c, TPB), TPB, 0, stream>>>(m1, NEG_INF, N * H1c);
    fill_f32<<<cdiv((long)N * H1c, TPB), TPB, 0, stream>>>(z1, 0.f, N * H1c);
    fill_f32<<<cdiv((long)N * HID_DIM, TPB), TPB, 0, stream>>>(acc1, 0.f, N * HID_DIM);

    edge_max<H1c><<<cdiv((long)ET * H1c, TPB), TPB, 0, stream>>>(ei, E, ET, es1, ed1, m1);
    edge_expsum<H1c><<<cdiv((long)ET * H1c, TPB), TPB, 0, stream>>>(ei, E, ET, es1, ed1, m1, p1, z1);
    edge_aggregate<H1c, C1c><<<cdiv((long)ET * (HID_DIM / 4), TPB), TPB, 0, stream>>>(
        ei, E, ET, h1, p1, z1, acc1);
    finish1<<<cdiv((long)N * HID_DIM, TPB), TPB, 0, stream>>>(acc1, b1, N, HID_DIM);
    // acc1 now holds h2 = elu(gat1 + b1)

    // ---------------- layer 2 ----------------
    {
        int waves = (N / 16) * (OUT_DIM / 64);
        gemm_wmma_f32_nb4<<<cdiv((long)waves * 32, TPB), TPB, 0, stream>>>(
            acc1, W2, t2, N, HID_DIM, OUT_DIM);
    }
    node_scores<H2c, C2c><<<cdiv((long)N * H2c, TPB), TPB, 0, stream>>>(
        t2, as2, ad2, es2, ed2, N);

    fill_f32<<<cdiv((long)N, TPB), TPB, 0, stream>>>(m2, NEG_INF, N);
    fill_f32<<<cdiv((long)N, TPB), TPB, 0, stream>>>(z2, 0.f, N);
    fill_f32<<<cdiv((long)N * OUT_DIM, TPB), TPB, 0, stream>>>(out, 0.f, N * OUT_DIM);

    edge_max<H2c><<<cdiv((long)ET * H2c, TPB), TPB, 0, stream>>>(ei, E, ET, es2, ed2, m2);
    edge_expsum<H2c><<<cdiv((long)ET * H2c, TPB), TPB, 0, stream>>>(ei, E, ET, es2, ed2, m2, p2, z2);
    edge_aggregate<H2c, C2c><<<cdiv((long)ET * (OUT_DIM / 4), TPB), TPB, 0, stream>>>(
        ei, E, ET, t2, p2, z2, out);
    finish2<<<cdiv((long)N * OUT_DIM, TPB), TPB, 0, stream>>>(out, b2, N, OUT_DIM);
}